// LossFunction_33689723469855
// MI455X (gfx1250) — compile-verified
//
#include <hip/hip_runtime.h>
#include <math.h>

typedef __attribute__((ext_vector_type(2))) float v2f;
typedef __attribute__((ext_vector_type(8))) float v8f;

#define NB 512      // batch
#define DD 512      // feature dim
#define CC 50000    // classes
#define KK 101      // top-k
#define EPSF 1e-8f

__device__ __forceinline__ unsigned f2key(float f) {
    unsigned u = __float_as_uint(f);
    return (u & 0x80000000u) ? ~u : (u | 0x80000000u);
}
__device__ __forceinline__ float key2f(unsigned k) {
    unsigned u = (k & 0x80000000u) ? (k & 0x7FFFFFFFu) : ~k;
    return __uint_as_float(u);
}
__device__ __forceinline__ float hard_sigmoid(float x) {
    return fminf(fmaxf(x + 3.0f, 0.0f), 6.0f) * (1.0f / 6.0f);
}

// ---------------------------------------------------------------------------
// Kernel 1: exact top-K mean/std per row of cosine (1024 rows of 50000),
// via 3-level radix select (11+11+10 bits) + one tie-exact sum pass.
// Row r = b*2 + p;  p==1 -> anchor stats, p==0 -> positive stats.
// ---------------------------------------------------------------------------
__global__ void topk_stats_kernel(const float* __restrict__ cosine,
                                  const float* __restrict__ w2p,
                                  const float* __restrict__ w3p,
                                  const float* __restrict__ b2p,
                                  const float* __restrict__ b3p,
                                  float* __restrict__ hm1, float* __restrict__ hd1,
                                  float* __restrict__ hm2, float* __restrict__ hd2) {
    const int row = blockIdx.x;
    const int tid = threadIdx.x;
    const float4* src = (const float4*)(cosine + (size_t)row * CC);
    const int NV = CC / 4;  // 12500 float4s, 16B-aligned (200000 % 16 == 0)

    __shared__ unsigned hist[2048];
    __shared__ unsigned s_chosen, s_above;
    __shared__ float s_red[256], s_red2[256];

    int kneed = KK;

    // ---- level 0: top 11 bits ----
    for (int i = tid; i < 2048; i += 256) hist[i] = 0;
    __syncthreads();
    for (int i = tid; i < NV; i += 256) {
        float4 v = src[i];
        atomicAdd(&hist[f2key(v.x) >> 21], 1u);
        atomicAdd(&hist[f2key(v.y) >> 21], 1u);
        atomicAdd(&hist[f2key(v.z) >> 21], 1u);
        atomicAdd(&hist[f2key(v.w) >> 21], 1u);
    }
    __syncthreads();
    if (tid == 0) {
        unsigned cum = 0;
        for (int i = 2047; i >= 0; --i) {
            unsigned c = hist[i];
            if (cum + c >= (unsigned)kneed) { s_chosen = (unsigned)i; s_above = cum; break; }
            cum += c;
        }
    }
    __syncthreads();
    const unsigned prefix11 = s_chosen;
    kneed -= (int)s_above;
    __syncthreads();

    // ---- level 1: next 11 bits within prefix11 ----
    for (int i = tid; i < 2048; i += 256) hist[i] = 0;
    __syncthreads();
    for (int i = tid; i < NV; i += 256) {
        float4 v = src[i];
        float c4[4] = {v.x, v.y, v.z, v.w};
        #pragma unroll
        for (int q = 0; q < 4; ++q) {
            unsigned k = f2key(c4[q]);
            if ((k >> 21) == prefix11) atomicAdd(&hist[(k >> 10) & 0x7FFu], 1u);
        }
    }
    __syncthreads();
    if (tid == 0) {
        unsigned cum = 0;
        for (int i = 2047; i >= 0; --i) {
            unsigned c = hist[i];
            if (cum + c >= (unsigned)kneed) { s_chosen = (unsigned)i; s_above = cum; break; }
            cum += c;
        }
    }
    __syncthreads();
    const unsigned prefix22 = (prefix11 << 11) | s_chosen;
    kneed -= (int)s_above;
    __syncthreads();

    // ---- level 2: last 10 bits within prefix22 ----
    for (int i = tid; i < 1024; i += 256) hist[i] = 0;
    __syncthreads();
    for (int i = tid; i < NV; i += 256) {
        float4 v = src[i];
        float c4[4] = {v.x, v.y, v.z, v.w};
        #pragma unroll
        for (int q = 0; q < 4; ++q) {
            unsigned k = f2key(c4[q]);
            if ((k >> 10) == prefix22) atomicAdd(&hist[k & 0x3FFu], 1u);
        }
    }
    __syncthreads();
    if (tid == 0) {
        unsigned cum = 0;
        for (int i = 1023; i >= 0; --i) {
            unsigned c = hist[i];
            if (cum + c >= (unsigned)kneed) { s_chosen = (unsigned)i; s_above = cum; break; }
            cum += c;
        }
    }
    __syncthreads();
    const unsigned T = (prefix22 << 10) | s_chosen;   // key of K-th largest
    kneed -= (int)s_above;                            // #ties of T inside top-K
    __syncthreads();

    // ---- final pass: sum / sumsq of strictly-greater elements ----
    float sum = 0.0f, sumsq = 0.0f;
    for (int i = tid; i < NV; i += 256) {
        float4 v = src[i];
        float c4[4] = {v.x, v.y, v.z, v.w};
        #pragma unroll
        for (int q = 0; q < 4; ++q) {
            float f = c4[q];
            if (f2key(f) > T) { sum += f; sumsq += f * f; }
        }
    }
    s_red[tid] = sum;
    s_red2[tid] = sumsq;
    __syncthreads();
    for (int s = 128; s > 0; s >>= 1) {
        if (tid < s) { s_red[tid] += s_red[tid + s]; s_red2[tid] += s_red2[tid + s]; }
        __syncthreads();
    }
    if (tid == 0) {
        float tv   = key2f(T);
        float tot  = s_red[0]  + (float)kneed * tv;
        float tot2 = s_red2[0] + (float)kneed * tv * tv;
        float mean = tot / (float)KK;
        float var  = (tot2 - (float)KK * mean * mean) / (float)(KK - 1);
        float sd   = sqrtf(fmaxf(var, 0.0f));
        float w2 = w2p[0], w3 = w3p[0], b2 = b2p[0], b3 = b3p[0];
        float hsm = hard_sigmoid(mean * w2 + w3);
        float hss = hard_sigmoid(sd * b2 + b3);
        int b = row >> 1, p = row & 1;
        if (p) { hm1[b] = hsm; hd1[b] = hss; }   // anchor (cosine[:,1,:])
        else   { hm2[b] = hsm; hd2[b] = hss; }   // positive (cosine[:,0,:])
    }
}

// ---------------------------------------------------------------------------
// Kernel 2: pn[b] = max(||x[b,0,:]||, EPS), an[b] = max(||x[b,1,:]||, EPS)
// ---------------------------------------------------------------------------
__global__ void norms_kernel(const float* __restrict__ x,
                             float* __restrict__ pn, float* __restrict__ an) {
    const int b = blockIdx.x, t = threadIdx.x;
    const float* xb = x + (size_t)b * (2 * DD);
    float a0 = xb[t],        a1 = xb[t + 256];
    float c0 = xb[DD + t],   c1 = xb[DD + t + 256];
    __shared__ float r0[256], r1[256];
    r0[t] = a0 * a0 + a1 * a1;
    r1[t] = c0 * c0 + c1 * c1;
    __syncthreads();
    for (int s = 128; s > 0; s >>= 1) {
        if (t < s) { r0[t] += r0[t + s]; r1[t] += r1[t + s]; }
        __syncthreads();
    }
    if (t == 0) {
        pn[b] = fmaxf(sqrtf(r0[0]), EPSF);
        an[b] = fmaxf(sqrtf(r1[0]), EPSF);
    }
}

// ---------------------------------------------------------------------------
// Kernel 3: out_dot via V_WMMA_F32_16X16X4_F32, fused normalization + logits.
// One wave per 16x16 tile; 1024 tiles; K-loop = 128 WMMAs.
// fp32 A layout: lane l<16 holds M=l, {K=k0,k0+1}; lane l>=16 holds M=l-16,
// {K=k0+2,k0+3}. B layout mirrors it (rows K across half-waves, N across lanes).
// ---------------------------------------------------------------------------
__global__ void gemm_logits_kernel(const float* __restrict__ x,
                                   const float* __restrict__ pn,
                                   const float* __restrict__ an,
                                   const float* __restrict__ hm1, const float* __restrict__ hd1,
                                   const float* __restrict__ hm2, const float* __restrict__ hd2,
                                   const float* __restrict__ wp, const float* __restrict__ bp,
                                   float* __restrict__ logits) {
    const int wid  = threadIdx.x >> 5;               // wave in block (0..7)
    const int w    = blockIdx.x * 8 + wid;           // global wave = tile id
    const int ti   = w >> 5;                         // tile row (0..31)
    const int tj   = w & 31;                         // tile col (0..31)
    const int I0   = ti * 16;
    const int J0   = tj * 16;
    const int lane = threadIdx.x & 31;
    const int hlf  = lane >> 4;                      // 0: K+{0,1}, 1: K+{2,3}
    const int lm   = lane & 15;

    // x[b][p][d] at b*1024 + p*512 + d
    const size_t a_base = (size_t)(I0 + lm) * (2 * DD);        // positive, p=0
    const size_t b_base = (size_t)(J0 + lm) * (2 * DD) + DD;   // anchor,   p=1

    v8f c = {};
    #pragma unroll 4
    for (int k0 = 0; k0 < DD; k0 += 4) {
        const int ka = k0 + 2 * hlf;
        v2f a  = *(const v2f*)(x + a_base + ka);
        v2f bb = *(const v2f*)(x + b_base + ka);
        c = __builtin_amdgcn_wmma_f32_16x16x4_f32(
                false, a, false, bb, (short)0, c, false, false);
    }

    const float wv = wp[0], bv = bp[0];
    const int j = J0 + lm;
    const float anj  = an[j];
    const float hm1j = hm1[j];
    const float hd1j = hd1[j];
    #pragma unroll
    for (int g = 0; g < 8; ++g) {
        const int i = I0 + g + 8 * hlf;              // C/D layout: lanes 16-31 hold M=g+8
        float val = c[g] / (pn[i] * anj);
        float d1 = (val - hm1j)   / hd1j;
        float d2 = (val - hm2[i]) / hd2[i];
        logits[(size_t)i * NB + j] = 0.5f * (d1 + d2) * wv + bv;
    }
}

// ---------------------------------------------------------------------------
// Kernel 4: per-row log-softmax diagonal term: rowloss[i] = logits[i,i] - lse(row i)
// ---------------------------------------------------------------------------
__global__ void row_logsoftmax_diag(const float* __restrict__ logits,
                                    float* __restrict__ rowloss) {
    const int i = blockIdx.x, t = threadIdx.x;
    const float* rowp = logits + (size_t)i * NB;
    float v0 = rowp[t], v1 = rowp[t + 256];
    __shared__ float red[256];
    __shared__ float s_diag;
    if (i < 256) { if (t == i) s_diag = v0; }
    else         { if (t == i - 256) s_diag = v1; }
    red[t] = fmaxf(v0, v1);
    __syncthreads();
    for (int s = 128; s > 0; s >>= 1) {
        if (t < s) red[t] = fmaxf(red[t], red[t + s]);
        __syncthreads();
    }
    const float m = red[0];
    __syncthreads();
    red[t] = expf(v0 - m) + expf(v1 - m);
    __syncthreads();
    for (int s = 128; s > 0; s >>= 1) {
        if (t < s) red[t] += red[t + s];
        __syncthreads();
    }
    if (t == 0) rowloss[i] = s_diag - m - logf(red[0]);
}

// ---------------------------------------------------------------------------
// Kernel 5: fixed-order tree reduction -> scalar loss (deterministic)
// ---------------------------------------------------------------------------
__global__ void final_reduce(const float* __restrict__ rowloss, float* __restrict__ out) {
    const int t = threadIdx.x;
    __shared__ float red[256];
    red[t] = rowloss[t] + rowloss[t + 256];
    __syncthreads();
    for (int s = 128; s > 0; s >>= 1) {
        if (t < s) red[t] += red[t + s];
        __syncthreads();
    }
    if (t == 0) out[0] = -red[0] / (float)NB;
}

extern "C" void kernel_launch(void* const* d_in, const int* in_sizes, int n_in,
                              void* d_out, int out_size, void* d_ws, size_t ws_size,
                              hipStream_t stream) {
    const float* x      = (const float*)d_in[0];   // (512,2,512)
    const float* cosine = (const float*)d_in[1];   // (512,2,50000)
    // d_in[2] = label (unused by the reference math)
    const float* wp  = (const float*)d_in[3];
    const float* bp  = (const float*)d_in[4];
    const float* w2p = (const float*)d_in[5];
    const float* w3p = (const float*)d_in[6];
    const float* b2p = (const float*)d_in[7];
    const float* b3p = (const float*)d_in[8];

    float* ws      = (float*)d_ws;
    float* logits  = ws;                       // 512*512
    float* hm1     = ws + (size_t)NB * NB;     // 512 each below
    float* hd1     = hm1 + NB;
    float* hm2     = hd1 + NB;
    float* hd2     = hm2 + NB;
    float* pn      = hd2 + NB;
    float* an      = pn + NB;
    float* rowloss = an + NB;

    topk_stats_kernel<<<NB * 2, 256, 0, stream>>>(cosine, w2p, w3p, b2p, b3p,
                                                  hm1, hd1, hm2, hd2);
    norms_kernel<<<NB, 256, 0, stream>>>(x, pn, an);
    gemm_logits_kernel<<<128, 256, 0, stream>>>(x, pn, an, hm1, hd1, hm2, hd2,
                                                wp, bp, logits);
    row_logsoftmax_diag<<<NB, 256, 0, stream>>>(logits, rowloss);
    final_reduce<<<1, 256, 0, stream>>>(rowloss, (float*)d_out);
}